// loss_function_893353197662
// MI455X (gfx1250) — compile-verified
//
#include <hip/hip_runtime.h>

// ---------------------------------------------------------------------------
// Gaussian NLL loss, C=3, per-pixel 3x3 SPD inverse in closed form.
// Memory-bound: 50.3 MB streamed once -> ~2.2 us at 23.3 TB/s. We use the
// CDNA5 async global->LDS pipeline (GLOBAL_LOAD_ASYNC_TO_LDS_B128, ASYNCcnt)
// with double buffering, then a deterministic two-stage reduction
// (ds_swizzle wave32 butterfly -> LDS -> per-block partial -> finalize).
// ---------------------------------------------------------------------------

namespace {

constexpr int   kN       = 16;
constexpr int   kHW      = 256 * 256;            // 65536 pixels per image
constexpr int   kNPix    = kN * kHW;             // 1,048,576
constexpr int   kQ       = 512;                  // pixels per tile
constexpr int   kTilesPerImg = kHW / kQ;         // 128
constexpr int   kTiles   = kNPix / kQ;           // 2048
constexpr int   kPlanes  = 12;                   // 9 output chans + 3 truth chans
constexpr int   kTileF   = kPlanes * kQ;         // 6144 floats per LDS buffer (24 KB)
constexpr int   kThreads = 256;                  // 8 waves (wave32)
constexpr int   kChunks  = (kPlanes * kQ) / (kThreads * 4); // 6 x b128 per thread
constexpr int   kBlocks  = 1024;                 // 2 tiles per block (double buffer)
constexpr float kS       = 625.00001f;           // sigma^2 + eps

// Exact parameter types of __builtin_amdgcn_global_load_async_to_lds_b128:
// (int4 addrspace(1)*, int4 addrspace(3)*, imm offset, imm cpol)
typedef int v4i __attribute__((__vector_size__(4 * sizeof(int))));
typedef __attribute__((address_space(1))) v4i gv4i;   // global int4
typedef __attribute__((address_space(3))) v4i lv4i;   // LDS int4

// One 16-byte async copy per lane: global -> LDS, tracked by ASYNCcnt.
// Low 32 bits of a flat LDS pointer are the wave-relative LDS byte address
// (ISA 10.2 aperture rules), so an integer round-trip is a valid AS cast.
__device__ __forceinline__ void async_cp16(const float* g, const float* l) {
  unsigned lds_addr = (unsigned)(size_t)l;
#if __has_builtin(__builtin_amdgcn_global_load_async_to_lds_b128)
  __builtin_amdgcn_global_load_async_to_lds_b128(
      (gv4i*)(size_t)g, (lv4i*)(size_t)lds_addr, 0, 0);
#else
  asm volatile("global_load_async_to_lds_b128 %0, %1, off"
               :: "v"(lds_addr), "v"(g) : "memory");
#endif
}

template <int N>
__device__ __forceinline__ void wait_async() {
#if __has_builtin(__builtin_amdgcn_s_wait_asynccnt)
  __builtin_amdgcn_s_wait_asynccnt(N);
#else
  asm volatile("s_wait_asynccnt %0" :: "i"(N));
#endif
}

// Wave32 butterfly sum via ds_swizzle (group-of-32 xor patterns).
__device__ __forceinline__ float wave_sum(float x) {
  union U { float f; int i; } u;
  u.f = x; u.i = __builtin_amdgcn_ds_swizzle(u.i, 0x401F); x += u.f; // SWAPX16
  u.f = x; u.i = __builtin_amdgcn_ds_swizzle(u.i, 0x201F); x += u.f; // SWAPX8
  u.f = x; u.i = __builtin_amdgcn_ds_swizzle(u.i, 0x101F); x += u.f; // SWAPX4
  u.f = x; u.i = __builtin_amdgcn_ds_swizzle(u.i, 0x081F); x += u.f; // SWAPX2
  u.f = x; u.i = __builtin_amdgcn_ds_swizzle(u.i, 0x041F); x += u.f; // SWAPX1
  return x;
}

} // namespace

__global__ void __launch_bounds__(kThreads)
nll_partial(const float* __restrict__ out9,   // (16, 9, 256, 256)
            const float* __restrict__ tru3,   // (16, 3, 256, 256)
            float* __restrict__ partial) {    // kBlocks partial sums
  __shared__ float lbuf[2 * kTileF];          // 48 KB double buffer
  __shared__ float wsum[kThreads / 32];

  const int tid = threadIdx.x;

  // Issue the 12 channel slabs of one tile into LDS buffer `bsel`.
  auto issue = [&](int tile, int bsel) {
    const int n  = tile >> 7;                 // tile / kTilesPerImg
    const int q0 = (tile & (kTilesPerImg - 1)) * kQ;
    float* dst = lbuf + bsel * kTileF;
#pragma unroll
    for (int i = 0; i < kChunks; ++i) {
      const int k     = i * kThreads + tid;   // 0..1535 16B-chunk index
      const int plane = k >> 7;               // 128 chunks per 512-float plane
      const int e4    = (k & 127) << 2;       // float offset within plane
      const float* src = (plane < 9)
          ? out9 + ((size_t)n * 9 + plane) * kHW + q0 + e4
          : tru3 + ((size_t)n * 3 + (plane - 9)) * kHW + q0 + e4;
      async_cp16(src, dst + plane * kQ + e4);
    }
  };

  float acc = 0.0f;
  int   buf = 0;
  int   t   = blockIdx.x;
  if (t < kTiles) issue(t, 0);

  for (; t < kTiles; t += gridDim.x) {
    const int  nt   = t + gridDim.x;
    const bool more = nt < kTiles;
    if (more) issue(nt, buf ^ 1);             // prefetch next tile into other buffer
    if (more) wait_async<kChunks>();          // oldest 6 (current tile) retired
    else      wait_async<0>();
    __syncthreads();                          // publish LDS across all 8 waves

    const float* lb = lbuf + buf * kTileF;
#pragma unroll
    for (int j = 0; j < kQ / kThreads; ++j) {
      const int px = tid + j * kThreads;      // lane-contiguous: bank-conflict free
      const float m0 = lb[ 0*kQ+px], m1 = lb[ 1*kQ+px], m2 = lb[ 2*kQ+px];
      const float a  = lb[ 3*kQ+px], b  = lb[ 4*kQ+px], c  = lb[ 5*kQ+px];
      const float d  = lb[ 6*kQ+px], e  = lb[ 7*kQ+px], f  = lb[ 8*kQ+px];
      const float d0 = lb[ 9*kQ+px] - m0;
      const float d1 = lb[10*kQ+px] - m1;
      const float d2 = lb[11*kQ+px] - m2;

      // A = [[a,b,c],[d,e,0],[f,0,0]]  (POS scatter);  V = A^T A + S*I
      const float V00 = fmaf(a, a, fmaf(d, d, fmaf(f, f, kS)));
      const float V01 = fmaf(a, b, d * e);
      const float V02 = a * c;
      const float V11 = fmaf(b, b, fmaf(e, e, kS));
      const float V12 = b * c;
      const float V22 = fmaf(c, c, kS);

      // Cofactors / determinant (symmetric 3x3)
      const float c00 = fmaf(V11, V22, -V12 * V12);
      const float c01 = fmaf(V02, V12, -V01 * V22);
      const float c02 = fmaf(V01, V12, -V02 * V11);
      const float det = fmaf(V00, c00, fmaf(V01, c01, V02 * c02));
      const float c11 = fmaf(V00, V22, -V02 * V02);
      const float c12 = fmaf(V01, V02, -V00 * V12);
      const float c22 = fmaf(V00, V11, -V01 * V01);

      // d^T adj(V) d
      const float q = fmaf(d0 * d0, c00,
                      fmaf(d1 * d1, c11,
                      fmaf(d2 * d2, c22,
                      2.0f * fmaf(d0 * d1, c01,
                             fmaf(d0 * d2, c02, d1 * d2 * c12)))));

      acc += 0.5f * (q / det + __logf(det));
    }
    __syncthreads();                          // all reads done before buffer reuse
    buf ^= 1;
  }

  // Block reduction: wave32 butterfly, then across the 8 waves.
  acc = wave_sum(acc);
  if ((tid & 31) == 0) wsum[tid >> 5] = acc;
  __syncthreads();
  if (tid == 0) {
    float s = 0.0f;
#pragma unroll
    for (int w = 0; w < kThreads / 32; ++w) s += wsum[w];
    partial[blockIdx.x] = s;                  // deterministic: fixed order, no atomics
  }
}

__global__ void __launch_bounds__(kThreads)
nll_finalize(const float* __restrict__ partial, float* __restrict__ loss) {
  __shared__ float wsum[kThreads / 32];
  const int tid = threadIdx.x;
  float s = 0.0f;
  for (int i = tid; i < kBlocks; i += kThreads) s += partial[i];
  s = wave_sum(s);
  if ((tid & 31) == 0) wsum[tid >> 5] = s;
  __syncthreads();
  if (tid == 0) {
    float tot = 0.0f;
#pragma unroll
    for (int w = 0; w < kThreads / 32; ++w) tot += wsum[w];
    loss[0] = tot / (float)kNPix - 2.5f;      // mean(likelihood) - 0.1*SIGMA
  }
}

extern "C" void kernel_launch(void* const* d_in, const int* in_sizes, int n_in,
                              void* d_out, int out_size, void* d_ws, size_t ws_size,
                              hipStream_t stream) {
  (void)in_sizes; (void)n_in; (void)out_size; (void)ws_size;
  const float* out9 = (const float*)d_in[0];  // (16,9,256,256) fp32
  const float* tru3 = (const float*)d_in[1];  // (16,3,256,256) fp32
  float* partial = (float*)d_ws;              // kBlocks floats of scratch
  float* loss    = (float*)d_out;             // scalar fp32

  nll_partial<<<kBlocks, kThreads, 0, stream>>>(out9, tru3, partial);
  nll_finalize<<<1, kThreads, 0, stream>>>(partial, loss);
}